// KPConv_12489764897253
// MI455X (gfx1250) — compile-verified
//
#include <hip/hip_runtime.h>
#include <math.h>
#include <float.h>

// ---- problem constants (match reference setup_inputs) ----
#define B_     2
#define N_     8192
#define M_     2048
#define CIN    64
#define COUT   128
#define NKP    15
#define MAXNB  32
#define RADIUS 0.5f
#define R2     (RADIUS * RADIUS)
#define SIGMA  (RADIUS * 0.3f)
#define INV2SIG2 (1.0f / (2.0f * SIGMA * SIGMA))
#define CHUNK  2048           // points staged in LDS per iteration (24KB)
#define KPAD   960            // 15 kernel points * 64 channels (GEMM K)
#define KCH    64             // GEMM K-chunk staged in LDS
#define AS_STRIDE 68          // 64 + 4 pad  -> A-fragment reads hit distinct banks
#define BS_STRIDE 132         // 128 + 4 pad -> B-fragment rows decorrelated

typedef __attribute__((ext_vector_type(2))) float v2f;
typedef __attribute__((ext_vector_type(4))) float v4f;
typedef __attribute__((ext_vector_type(8))) float v8f;
typedef int v4i_gcc __attribute__((__vector_size__(16)));   // matches builtin param

// ---------------------------------------------------------------------------
// gfx1250 async global->LDS copy (ASYNCcnt path), with portable fallback.
// Builtin signature (from hipcc diagnostic): (v4i AS1*, v4i AS3*, Ii, Ii)
// ---------------------------------------------------------------------------
__device__ __forceinline__ void cp16_async(void* lds, const void* glb) {
#if defined(__has_builtin) && __has_builtin(__builtin_amdgcn_global_load_async_to_lds_b128)
    __builtin_amdgcn_global_load_async_to_lds_b128(
        (__attribute__((address_space(1))) v4i_gcc*)(uintptr_t)glb,
        (__attribute__((address_space(3))) v4i_gcc*)(uint32_t)(uintptr_t)lds,
        0, 0);
#else
    *(v4f*)lds = *(const v4f*)glb;
#endif
}

__device__ __forceinline__ void async_wait0() {
#if defined(__has_builtin) && __has_builtin(__builtin_amdgcn_s_wait_asynccnt)
    __builtin_amdgcn_s_wait_asynccnt(0);
#else
    asm volatile("s_wait_asynccnt 0x0" ::: "memory");
#endif
}

// ---------------------------------------------------------------------------
// wave32 compare-exchange on (dist, idx) pairs across lanes (bitonic network)
// ---------------------------------------------------------------------------
__device__ __forceinline__ void ce_pair(float& d, int& i, int j, bool asc) {
    float od = __shfl_xor(d, j, 32);
    int   oi = __shfl_xor(i, j, 32);
    int lane = threadIdx.x & 31;
    bool lower    = (lane & j) == 0;
    bool keep_min = (lower == asc);
    bool take = keep_min ? (od < d) : (od > d);
    if (take) { d = od; i = oi; }
}

__device__ __forceinline__ void bitonic_sort32(float& d, int& i, bool asc) {
    int lane = threadIdx.x & 31;
#pragma unroll
    for (int k = 2; k <= 32; k <<= 1) {
#pragma unroll
        for (int j = k >> 1; j > 0; j >>= 1) {
            bool dir = (((lane & k) == 0) == asc);
            ce_pair(d, i, j, dir);
        }
    }
}

// ---------------------------------------------------------------------------
// Kernel 1: ball query (top-32 smallest d^2 within radius) + Gaussian weights
// One wave per query; lane l holds slot l of the ascending-sorted best-32.
// ---------------------------------------------------------------------------
extern "C" __global__ void __launch_bounds__(256)
kpconv_ballquery(const float* __restrict__ points,
                 const float* __restrict__ query,
                 const float* __restrict__ kp,
                 int*   __restrict__ idx_ws,     // (B*M, 32)
                 float* __restrict__ w_ws)       // (B*M, 32, 16) k padded to 16
{
    __shared__ float spts[CHUNK * 3];
    const int lane = threadIdx.x & 31;
    const int wave = threadIdx.x >> 5;
    const int b  = blockIdx.x / (M_ / 8);
    const int m  = (blockIdx.x % (M_ / 8)) * 8 + wave;
    const int bm = b * M_ + m;

    const float qx = query[bm * 3 + 0];
    const float qy = query[bm * 3 + 1];
    const float qz = query[bm * 3 + 2];

    float sd = FLT_MAX;   // slot distance^2 (ascending across lanes)
    int   si = 0;         // slot point index

    for (int ch = 0; ch < N_; ch += CHUNK) {
        __syncthreads();
        for (int t = threadIdx.x; t < CHUNK * 3; t += 256)
            spts[t] = points[(size_t)(b * N_ + ch) * 3 + t];
        __syncthreads();

        for (int g = 0; g < CHUNK; g += 32) {
            const int pl = g + lane;
            float dx = spts[pl * 3 + 0] - qx;
            float dy = spts[pl * 3 + 1] - qy;
            float dz = spts[pl * 3 + 2] - qz;
            float d2 = dx * dx + dy * dy + dz * dz;
            float cd = (d2 <= R2) ? d2 : FLT_MAX;  // outside radius -> never selected
            int   ci = ch + pl;

            // wave-uniform early skip: can this chunk-group beat the 32nd best?
            float gm = cd;
#pragma unroll
            for (int o = 16; o > 0; o >>= 1) gm = fminf(gm, __shfl_xor(gm, o, 32));
            float kth = __shfl(sd, 31, 32);
            if (gm < kth) {
                bitonic_sort32(cd, ci, /*asc=*/false);      // incoming descending
                if (cd < sd) { sd = cd; si = ci; }          // pairwise min -> best 32 (bitonic)
#pragma unroll
                for (int j = 16; j > 0; j >>= 1)            // cleanup -> ascending
                    ce_pair(sd, si, j, true);
            }
        }
    }

    const bool valid = (sd <= R2);
    int ii = valid ? si : 0;
    if (ii < 0 || ii >= N_) ii = 0;
    idx_ws[bm * 32 + lane] = ii;

    const float px = points[(size_t)(b * N_ + ii) * 3 + 0];
    const float py = points[(size_t)(b * N_ + ii) * 3 + 1];
    const float pz = points[(size_t)(b * N_ + ii) * 3 + 2];
    const float rx = px - qx, ry = py - qy, rz = pz - qz;

    float* wq = w_ws + (size_t)bm * (MAXNB * 16) + lane * 16;
#pragma unroll
    for (int k = 0; k < NKP; ++k) {
        float ex = rx - kp[k * 3 + 0];
        float ey = ry - kp[k * 3 + 1];
        float ez = rz - kp[k * 3 + 2];
        float sq = ex * ex + ey * ey + ez * ez;
        wq[k] = valid ? __expf(-sq * INV2SIG2) : 0.0f;
    }
    wq[15] = 0.0f;  // padded kernel point
}

// ---------------------------------------------------------------------------
// Kernel 2: Fk[bm, k, c] = sum_j w[bm,j,k] * feats[b, idx[bm,j], c]
// 64 threads (channels) per query; coalesced feature gathers.
// ---------------------------------------------------------------------------
extern "C" __global__ void __launch_bounds__(256)
kpconv_gather(const float* __restrict__ feats,
              const int*   __restrict__ idx_ws,
              const float* __restrict__ w_ws,
              float* __restrict__ fk_ws)          // (B*M, 960)
{
    const int c  = threadIdx.x & 63;
    const int ql = threadIdx.x >> 6;
    const int bm = blockIdx.x * 4 + ql;
    const int b  = bm / M_;

    float acc[NKP];
#pragma unroll
    for (int k = 0; k < NKP; ++k) acc[k] = 0.0f;

    const int*   idxq = idx_ws + bm * 32;
    const float* wq   = w_ws + (size_t)bm * (MAXNB * 16);
    for (int j = 0; j < MAXNB; ++j) {
        const int i = idxq[j];
        const float f = feats[((size_t)b * N_ + i) * CIN + c];
#pragma unroll
        for (int k = 0; k < NKP; ++k) acc[k] += wq[j * 16 + k] * f;
    }
    float* out = fk_ws + (size_t)bm * KPAD;
#pragma unroll
    for (int k = 0; k < NKP; ++k) out[k * CIN + c] = acc[k];
}

// ---------------------------------------------------------------------------
// Kernel 3: out (4096x128) = Fk (4096x960) * W (960x128) + bias, f32 WMMA.
// Block tile 64x128, 8 waves, 4 accumulator tiles each. Operands staged in
// LDS per 64-wide K-chunk via gfx1250 async global->LDS copies; fragments are
// served from LDS (padded strides avoid bank conflicts on fragment reads).
// A fragment (16x4): lanes 0-15 hold K{kk,kk+1}, lanes 16-31 K{kk+2,kk+3}.
// ---------------------------------------------------------------------------
extern "C" __global__ void __launch_bounds__(256)
kpconv_gemm(const float* __restrict__ A,     // fk_ws, (4096 x 960)
            const float* __restrict__ Wm,    // weight, (960 x 128) row-major
            const float* __restrict__ bias,  // (128)
            float* __restrict__ out)         // (4096 x 128)
{
    __shared__ float As[64 * AS_STRIDE];     // 17408 B
    __shared__ float Bs[KCH * BS_STRIDE];    // 33792 B

    const int tid  = threadIdx.x;
    const int lane = tid & 31;
    const int wave = tid >> 5;
    const int rblk = blockIdx.x * 64;
    const int r0 = (wave & 3) * 16;          // tile row base within block
    const int c0 = (wave >> 2) * 64;         // tile col base within block
    const int lm = lane & 15;                // row (A) / col (B) within tile
    const int lk = (lane >> 4) * 2;          // K sub-offset per lane half

    v8f acc[4] = {v8f{}, v8f{}, v8f{}, v8f{}};

    for (int kk = 0; kk < KPAD; kk += KCH) {
        __syncthreads();                     // LDS safe to overwrite

        // stage A chunk: 64 rows x 64 floats (16B granules)
#pragma unroll
        for (int i = 0; i < 4; ++i) {
            const int t   = tid + i * 256;
            const int row = t >> 4;
            const int c4  = (t & 15) * 4;
            cp16_async(&As[row * AS_STRIDE + c4],
                       A + (size_t)(rblk + row) * KPAD + kk + c4);
        }
        // stage B chunk: 64 rows x 128 floats
#pragma unroll
        for (int i = 0; i < 8; ++i) {
            const int t   = tid + i * 256;
            const int row = t >> 5;
            const int c4  = (t & 31) * 4;
            cp16_async(&Bs[row * BS_STRIDE + c4],
                       Wm + (size_t)(kk + row) * COUT + c4);
        }
        async_wait0();
        __syncthreads();                     // LDS visible to all waves

#pragma unroll
        for (int ks = 0; ks < KCH; ks += 4) {
            const v2f a = *(const v2f*)(&As[(r0 + lm) * AS_STRIDE + ks + lk]);
#pragma unroll
            for (int t = 0; t < 4; ++t) {
                const int col = c0 + t * 16 + lm;
                v2f bfrag;
                bfrag.x = Bs[(ks + lk)     * BS_STRIDE + col];
                bfrag.y = Bs[(ks + lk + 1) * BS_STRIDE + col];
                acc[t] = __builtin_amdgcn_wmma_f32_16x16x4_f32(
                    false, a, false, bfrag, (short)0, acc[t], false, false);
            }
        }
    }

#pragma unroll
    for (int t = 0; t < 4; ++t) {
        const int col = c0 + t * 16 + lm;
        const float bv = bias[col];
        const int rbase = rblk + r0 + (lane >> 4) * 8;
#pragma unroll
        for (int v = 0; v < 8; ++v)
            out[(size_t)(rbase + v) * COUT + col] = acc[t][v] + bv;
    }
}

// ---------------------------------------------------------------------------
extern "C" void kernel_launch(void* const* d_in, const int* in_sizes, int n_in,
                              void* d_out, int out_size, void* d_ws, size_t ws_size,
                              hipStream_t stream) {
    const float* points = (const float*)d_in[0];
    const float* feats  = (const float*)d_in[1];
    const float* query  = (const float*)d_in[2];
    const float* weight = (const float*)d_in[3];   // (15,64,128) == (960,128)
    const float* bias   = (const float*)d_in[4];
    const float* kp     = (const float*)d_in[5];

    char* ws = (char*)d_ws;
    int*   idx_ws = (int*)ws;                                   // 4096*32*4   = 512 KB
    float* w_ws   = (float*)(ws + (size_t)(1 << 19));           // 4096*512*4  =   8 MB
    float* fk_ws  = (float*)(ws + (size_t)(1 << 19) + (size_t)4096 * 512 * 4); // ~15.7 MB

    kpconv_ballquery<<<(B_ * M_) / 8, 256, 0, stream>>>(points, query, kp, idx_ws, w_ws);
    kpconv_gather  <<<(B_ * M_) / 4, 256, 0, stream>>>(feats, idx_ws, w_ws, fk_ws);
    kpconv_gemm    <<<(B_ * M_) / 64, 256, 0, stream>>>(fk_ws, weight, bias, (float*)d_out);
}